// RGCNLayer_58420145160623
// MI455X (gfx1250) — compile-verified
//
#include <hip/hip_runtime.h>
#include <math.h>

// RGCN layer for MI455X (gfx1250, wave32).
// messages = norm * sum_r adj[r] @ (H @ Wrel[r])   (associativity fusion)
// update   = H @ W0 + messages
// gate     = sigmoid([update, H] @ Wg)
// out      = tanh(update)*gate + H*(1-gate)
//
// Main pass streams adj (2.415 GB) via TDM tensor_load_to_lds (double
// buffered) and consumes it with v_wmma_f32_16x16x4_f32.

typedef __attribute__((ext_vector_type(2))) float v2f;
typedef __attribute__((ext_vector_type(8))) float v8f;
typedef unsigned int u32;
typedef __attribute__((ext_vector_type(4))) u32 u32x4;
typedef __attribute__((ext_vector_type(4))) int  i32x4;
typedef __attribute__((ext_vector_type(8))) int  i32x8;

#define N_NODES 12288
#define RELS 4
#define HD 64
#define KCH (N_NODES / 64)      // k-chunks of 64 per relation
#define ROWS_PER_WAVE 32

// ---------------------------------------------------------------------------
// TDM: load a 2D f32 tile (tile_k contiguous elems x tile_rows rows, row
// stride row_stride elems) from global to LDS. D# per CDNA5 ISA §8.3/8.4.
// Tile is treated as a self-contained tensor (tiles never cross bounds).
// ---------------------------------------------------------------------------
__device__ __forceinline__ void tdm_load_tile_f32(const float* gsrc,
                                                  u32 lds_byte_off,
                                                  u32 tile_k, u32 tile_rows,
                                                  u32 row_stride) {
  unsigned long long ga = (unsigned long long)(uintptr_t)gsrc;
  // Group 0: count=1 (valid user desc), lds_addr, global_addr[56:0], type=2
  u32x4 g0 = { 1u,
               lds_byte_off,
               (u32)ga,
               (u32)((ga >> 32) & 0x01ffffffu) | (2u << 30) };
  // Group 1: data_size=2 (4B), tensor_dim0/1 == tile dims, strides
  i32x8 g1 = { (int)(2u << 16),                                   // w0
               (int)((tile_k & 0xffffu) << 16),                   // w1: td0[15:0]
               (int)((tile_k >> 16) | ((tile_rows & 0xffffu) << 16)), // w2
               (int)((tile_rows >> 16) | (tile_k << 16)),         // w3: tile_dim0
               (int)tile_rows,                                    // w4: tile_dim1
               (int)row_stride,                                   // w5: stride[31:0]
               0,                                                 // w6
               0 };                                               // w7
  i32x4 gz = { 0, 0, 0, 0 };
#if defined(__clang_major__) && __clang_major__ >= 23
  i32x8 gz8 = { 0, 0, 0, 0, 0, 0, 0, 0 };
  __builtin_amdgcn_tensor_load_to_lds(g0, g1, gz, gz, gz8, 0);
#else
  __builtin_amdgcn_tensor_load_to_lds(g0, g1, gz, gz, 0);
#endif
}

__device__ __forceinline__ u32 lds_off_of(const void* p) {
  // Flat LDS-aperture address truncated to 32 bits == LDS byte offset.
  return (u32)(uintptr_t)p;
}

// ---------------------------------------------------------------------------
// Kernel 1: HW[r] = H @ Wrel[r]  (r=0..3),  U0 = H @ W0
// One wave per 16-row tile per output matrix. WMMA f32 16x16x4.
// Fragment layouts (ISA 7.12.2, 32-bit A 16x4 / B 4x16):
//   A: lane L<16: v0=A[M=L,K=k0], v1=A[M=L,K=k0+1]; L>=16: K=k0+2,k0+3
//   B: lane L<16: v0=B[K=k0,N=L], v1=B[K=k0+1,N=L]; L>=16: K=k0+2,k0+3
//   C/D: VGPR j: rows M=j (lanes 0-15), M=j+8 (lanes 16-31), N=lane&15
// ---------------------------------------------------------------------------
__global__ __launch_bounds__(32) void rgcn_prep(const float* __restrict__ H,
                                                const float* __restrict__ W0,
                                                const float* __restrict__ Wrel,
                                                float* __restrict__ HW,
                                                float* __restrict__ U0) {
  const int lane = threadIdx.x;
  const int row0 = blockIdx.x * 16;
  const int mat  = blockIdx.y;             // 0..3 -> Wrel[mat], 4 -> W0
  const float* __restrict__ W    = (mat < 4) ? (Wrel + (size_t)mat * HD * HD) : W0;
  float* __restrict__       outp = (mat < 4) ? (HW + (size_t)mat * N_NODES * HD) : U0;

  const int mrow = lane & 15;
  const int koff = (lane >> 4) << 1;       // 0 or 2
  const float* __restrict__ arow = H + (size_t)(row0 + mrow) * HD + koff;

  v8f acc[4] = {};
#pragma unroll
  for (int k0 = 0; k0 < HD; k0 += 4) {
    v2f a = *(const v2f*)(arow + k0);
    const float* bp = W + (size_t)(k0 + koff) * HD + mrow;
#pragma unroll
    for (int t = 0; t < 4; ++t) {
      v2f b;
      b.x = bp[t * 16];
      b.y = bp[t * 16 + HD];
      acc[t] = __builtin_amdgcn_wmma_f32_16x16x4_f32(false, a, false, b,
                                                     (short)0, acc[t], false, false);
    }
  }

  const int mhi = (lane >> 4) << 3;
#pragma unroll
  for (int t = 0; t < 4; ++t)
#pragma unroll
    for (int j = 0; j < 8; ++j)
      outp[(size_t)(row0 + j + mhi) * HD + t * 16 + mrow] = acc[t][j];
}

// ---------------------------------------------------------------------------
// Kernel 2: TDM-staged streaming pass over adj + fused epilogue.
// One wave owns 32 output rows (2 M-tiles x 64 cols). Stages of 32x64 f32
// (8 KB) are DMA'd into double-buffered LDS; B frags come from L2-resident
// HW. 768 stages per wave (4 relations x 192 k-chunks).
// ---------------------------------------------------------------------------
__global__ __launch_bounds__(32) void rgcn_main(const float* __restrict__ H,
                                                const float* __restrict__ adj,
                                                const float* __restrict__ Wg,
                                                const float* __restrict__ HW,
                                                const float* __restrict__ U0,
                                                float* __restrict__ out) {
  __shared__ float s_a[2][ROWS_PER_WAVE * 64];    // TDM staging, 2 x 8 KB
  __shared__ float s_upd[ROWS_PER_WAVE * HD];     // 8 KB
  __shared__ float s_h[ROWS_PER_WAVE * HD];       // 8 KB

  const int lane = threadIdx.x;
  const int row0 = blockIdx.x * ROWS_PER_WAVE;
  const float norm = 1.0f / (float)(N_NODES - 1);

  // Stage this tile's H rows into LDS (used by the gate epilogue).
  for (int i = lane; i < ROWS_PER_WAVE * HD; i += 32)
    s_h[i] = H[(size_t)row0 * HD + i];

  const int mrow = lane & 15;
  const int koff = (lane >> 4) << 1;       // 0 or 2
  const int mhi  = (lane >> 4) << 3;       // 0 or 8

  v8f acc[2][4] = {};

  const int NSTAGE = RELS * KCH;           // 768 stages of 32 rows x 64 k

  // Prologue: DMA stage 0.
  {
    const float* g0p = adj + ((size_t)0 * N_NODES + (size_t)row0) * N_NODES + 0;
    tdm_load_tile_f32(g0p, lds_off_of(&s_a[0][0]), 64, ROWS_PER_WAVE, N_NODES);
  }

  for (int s = 0; s < NSTAGE; ++s) {
    if (s + 1 < NSTAGE) {
      const int sn = s + 1;
      const int rn = sn / KCH;
      const int kkn = (sn % KCH) * 64;
      const float* gp =
          adj + ((size_t)rn * N_NODES + (size_t)row0) * N_NODES + kkn;
      // Ensure prior ds reads of the buffer being overwritten have retired.
      asm volatile("s_wait_dscnt 0" ::: "memory");
      tdm_load_tile_f32(gp, lds_off_of(&s_a[sn & 1][0]), 64, ROWS_PER_WAVE,
                        N_NODES);
      __builtin_amdgcn_s_wait_tensorcnt(1);   // oldest (stage s) complete
    } else {
      __builtin_amdgcn_s_wait_tensorcnt(0);
    }
    asm volatile("" ::: "memory");            // keep LDS reads below the wait

    const float* sa  = &s_a[s & 1][0];
    const int kkg    = (s % KCH) * 64;        // global k of this stage
    const float* hwr = HW + (size_t)(s / KCH) * N_NODES * HD;

#pragma unroll
    for (int k0 = 0; k0 < 64; k0 += 4) {
      v2f a0 = *(const v2f*)(sa + mrow * 64 + k0 + koff);          // ds_load_b64
      v2f a1 = *(const v2f*)(sa + (16 + mrow) * 64 + k0 + koff);
      const float* bp = hwr + (size_t)(kkg + k0 + koff) * HD + mrow;
#pragma unroll
      for (int t = 0; t < 4; ++t) {
        v2f b;
        b.x = bp[t * 16];
        b.y = bp[t * 16 + HD];
        acc[0][t] = __builtin_amdgcn_wmma_f32_16x16x4_f32(
            false, a0, false, b, (short)0, acc[0][t], false, false);
        acc[1][t] = __builtin_amdgcn_wmma_f32_16x16x4_f32(
            false, a1, false, b, (short)0, acc[1][t], false, false);
      }
    }
  }

  // update = norm*messages + U0  -> LDS tile (32 x 64)
#pragma unroll
  for (int mt = 0; mt < 2; ++mt)
#pragma unroll
    for (int t = 0; t < 4; ++t)
#pragma unroll
      for (int j = 0; j < 8; ++j) {
        const int m = mt * 16 + j + mhi;
        const int n = t * 16 + mrow;
        s_upd[m * HD + n] =
            acc[mt][t][j] * norm + U0[(size_t)(row0 + m) * HD + n];
      }
  __syncthreads();

  // Gate + blend epilogue: 2048 outputs / 32 lanes = 64 per lane.
  for (int idx = lane; idx < ROWS_PER_WAVE * HD; idx += 32) {
    const int m = idx >> 6;
    const int n = idx & 63;
    const float u = s_upd[idx];
    float g = 0.0f;
#pragma unroll 8
    for (int k = 0; k < HD; ++k)
      g = fmaf(s_upd[m * HD + k], Wg[k * HD + n], g);
#pragma unroll 8
    for (int k = 0; k < HD; ++k)
      g = fmaf(s_h[m * HD + k], Wg[(HD + k) * HD + n], g);
    g = 1.0f / (1.0f + expf(-g));
    out[(size_t)row0 * HD + idx] = tanhf(u) * g + s_h[idx] * (1.0f - g);
  }
}

// ---------------------------------------------------------------------------
extern "C" void kernel_launch(void* const* d_in, const int* in_sizes, int n_in,
                              void* d_out, int out_size, void* d_ws, size_t ws_size,
                              hipStream_t stream) {
  (void)in_sizes; (void)n_in; (void)out_size; (void)ws_size;
  const float* H    = (const float*)d_in[0];   // [N,64]
  const float* adj  = (const float*)d_in[1];   // [R,N,N]
  const float* W0   = (const float*)d_in[2];   // [64,64]
  const float* Wrel = (const float*)d_in[3];   // [R,64,64]
  const float* Wg   = (const float*)d_in[4];   // [128,64]
  float* out = (float*)d_out;                  // [N,64]

  // Workspace: HW [R,N,64] + U0 [N,64]  (~15.7 MB)
  float* HW = (float*)d_ws;
  float* U0 = HW + (size_t)RELS * N_NODES * HD;

  rgcn_prep<<<dim3(N_NODES / 16, 5), 32, 0, stream>>>(H, W0, Wrel, HW, U0);
  rgcn_main<<<dim3(N_NODES / ROWS_PER_WAVE), 32, 0, stream>>>(H, adj, Wg, HW,
                                                              U0, out);
}